// Aggregator_44435731644653
// MI455X (gfx1250) — compile-verified
//
#include <hip/hip_runtime.h>

typedef __attribute__((ext_vector_type(2))) float v2f;
typedef __attribute__((ext_vector_type(8))) float v8f;

#define D_DIM 512
#define CHUNKS 16

// ---------------------------------------------------------------------------
// Kernel 1: zero the fp32 accumulator and build the bag prefix-sum (B=64).
// ---------------------------------------------------------------------------
__global__ void bag_init_kernel(const int* __restrict__ counts, int B,
                                float* __restrict__ acc, int* __restrict__ prefix) {
    const int n = B * D_DIM;
    for (int i = threadIdx.x; i < n; i += blockDim.x) acc[i] = 0.0f;
    if (threadIdx.x == 0) {
        int s = 0;
        prefix[0] = 0;
        for (int i = 0; i < B; ++i) { s += counts[i]; prefix[i + 1] = s; }
    }
}

// ---------------------------------------------------------------------------
// Kernel 2: segmented row-sum via V_WMMA_F32_16X16X4_F32 (A = ones).
//   D[m,n] = sum_{k<4} B[k,n] + C[m,n]  -> folds 4 rows x 16 cols per WMMA.
// 8 waves x 4 column tiles = 512 columns. Each block owns a contiguous row
// slice of exactly one bag; all WMMAs execute at full EXEC.
// Main loop is branch-free with unconditional loads off one incremented base
// pointer (immediate offsets only); the ragged tail is handled once with
// clamped addresses + selects (no exec-mask branches).
// ---------------------------------------------------------------------------
__global__ void __launch_bounds__(256)
bag_sum_wmma_kernel(const float* __restrict__ samples,
                    const int* __restrict__ prefix,
                    float* __restrict__ acc) {
    const int bag   = blockIdx.y;
    const int start = prefix[bag];
    const int end   = prefix[bag + 1];
    const int count = end - start;
    const int rowsPer = (count + (int)gridDim.x - 1) / (int)gridDim.x;
    const int r0 = start + (int)blockIdx.x * rowsPer;
    const int r1 = min(r0 + rowsPer, end);
    if (r0 >= r1) return;                      // block-uniform exit

    const int lane  = threadIdx.x & 31;
    const int wave  = threadIdx.x >> 5;        // 0..7
    const int lmod  = lane & 15;               // column within 16-wide tile
    const int khalf = (lane >> 4) << 1;        // lanes 0-15: K=0/1, 16-31: K=2/3
    const int colBase = wave * 64;             // 4 tiles of 16 columns per wave

    v2f a; a.x = 1.0f; a.y = 1.0f;             // all-ones A matrix
    v8f c0 = {}, c1 = {}, c2 = {}, c3 = {};

    const int span  = r1 - r0;
    const int nfull = span >> 2;               // iterations with all 4 rows valid
    const int rem   = span & 3;

    // base pointer for this lane: row (r0 + khalf), its pair row is +D_DIM
    const float* p = samples + (size_t)(r0 + khalf) * D_DIM + colBase + lmod;

    #pragma unroll 2
    for (int it = 0; it < nfull; ++it) {
        __builtin_prefetch(p + 16 * D_DIM, 0, 3);   // ~32 KB ahead

        v2f b0, b1, b2, b3;
        b0.x = p[0];            b0.y = p[D_DIM];
        b1.x = p[16];           b1.y = p[D_DIM + 16];
        b2.x = p[32];           b2.y = p[D_DIM + 32];
        b3.x = p[48];           b3.y = p[D_DIM + 48];
        p += 4 * D_DIM;

        c0 = __builtin_amdgcn_wmma_f32_16x16x4_f32(false, a, false, b0, (short)0, c0, false, false);
        c1 = __builtin_amdgcn_wmma_f32_16x16x4_f32(false, a, false, b1, (short)0, c1, false, false);
        c2 = __builtin_amdgcn_wmma_f32_16x16x4_f32(false, a, false, b2, (short)0, c2, false, false);
        c3 = __builtin_amdgcn_wmma_f32_16x16x4_f32(false, a, false, b3, (short)0, c3, false, false);
    }

    if (rem) {                                  // block-uniform branch
        const int rbase = r1 - rem;             // first tail row (valid)
        const bool ok0 = khalf < rem;
        const bool ok1 = (khalf + 1) < rem;
        const int  o0  = ok0 ? khalf : 0;       // clamp to a valid row
        const int  o1  = ok1 ? (khalf + 1) : 0;
        const float* q0 = samples + (size_t)(rbase + o0) * D_DIM + colBase + lmod;
        const float* q1 = samples + (size_t)(rbase + o1) * D_DIM + colBase + lmod;

        v2f b0, b1, b2, b3;
        float x;
        x = q0[0];  b0.x = ok0 ? x : 0.0f;  x = q1[0];  b0.y = ok1 ? x : 0.0f;
        x = q0[16]; b1.x = ok0 ? x : 0.0f;  x = q1[16]; b1.y = ok1 ? x : 0.0f;
        x = q0[32]; b2.x = ok0 ? x : 0.0f;  x = q1[32]; b2.y = ok1 ? x : 0.0f;
        x = q0[48]; b3.x = ok0 ? x : 0.0f;  x = q1[48]; b3.y = ok1 ? x : 0.0f;

        c0 = __builtin_amdgcn_wmma_f32_16x16x4_f32(false, a, false, b0, (short)0, c0, false, false);
        c1 = __builtin_amdgcn_wmma_f32_16x16x4_f32(false, a, false, b1, (short)0, c1, false, false);
        c2 = __builtin_amdgcn_wmma_f32_16x16x4_f32(false, a, false, b2, (short)0, c2, false, false);
        c3 = __builtin_amdgcn_wmma_f32_16x16x4_f32(false, a, false, b3, (short)0, c3, false, false);
    }

    // Every lane holds a duplicate column sum in c*[0]:
    //   lanes 0-15 publish tiles 0,1 ; lanes 16-31 publish tiles 2,3.
    // Full-EXEC atomics, 2 per lane.
    const int  hi  = lane >> 4;                 // 0 or 1
    const float s0 = hi ? c2[0] : c0[0];
    const float s1 = hi ? c3[0] : c1[0];
    float* dst = acc + (size_t)bag * D_DIM + colBase + hi * 32 + lmod;
    atomicAdd(dst,      s0);
    atomicAdd(dst + 16, s1);
}

// ---------------------------------------------------------------------------
// Kernel 3: mean = sum / count
// ---------------------------------------------------------------------------
__global__ void bag_finalize_kernel(const float* __restrict__ acc,
                                    const int* __restrict__ counts,
                                    float* __restrict__ out, int n) {
    const int i = blockIdx.x * blockDim.x + threadIdx.x;
    if (i < n) {
        const int bag = i / D_DIM;
        out[i] = acc[i] / (float)counts[bag];
    }
}

extern "C" void kernel_launch(void* const* d_in, const int* in_sizes, int n_in,
                              void* d_out, int out_size, void* d_ws, size_t ws_size,
                              hipStream_t stream) {
    const float* samples = (const float*)d_in[0];
    const int*   counts  = (const int*)d_in[1];
    const int B = in_sizes[1];

    float* acc    = (float*)d_ws;
    int*   prefix = (int*)((char*)d_ws + (size_t)B * D_DIM * sizeof(float));
    float* out    = (float*)d_out;

    bag_init_kernel<<<dim3(1), dim3(256), 0, stream>>>(counts, B, acc, prefix);

    bag_sum_wmma_kernel<<<dim3(CHUNKS, B), dim3(256), 0, stream>>>(samples, prefix, acc);

    const int n = B * D_DIM;
    bag_finalize_kernel<<<dim3((n + 255) / 256), dim3(256), 0, stream>>>(acc, counts, out, n);
}